// GPRGNN_26723286516071
// MI455X (gfx1250) — compile-verified
//
#include <hip/hip_runtime.h>
#include <math.h>

typedef __attribute__((ext_vector_type(2))) float v2f;
typedef __attribute__((ext_vector_type(8))) float v8f;

constexpr int FIN = 128;   // input features
constexpr int HD  = 16;    // hidden
constexpr int NC  = 40;    // classes
constexpr int KHOPS = 10;
constexpr float AL  = 0.1f;
constexpr float EPS = 1e-5f;

// ---------------- init: zero accumulators, self-loop degree = 1 ----------------
__global__ void init_kernel(float* dinv, float* agg1, float* outacc, double* stats, int N) {
    unsigned i = blockIdx.x * blockDim.x + threadIdx.x;
    if (i < (unsigned)N * NC) outacc[i] = 0.f;
    if (i < (unsigned)N * HD) agg1[i] = 0.f;
    if (i < (unsigned)N)      dinv[i] = 1.0f;   // self-loop contribution to degree
    if (i < 32)               stats[i] = 0.0;
}

// ---------------- degree over dst, then dinv = rsqrt(deg) ----------------
__global__ void deg_kernel(const int* __restrict__ dst, float* deg, int E) {
    unsigned e = blockIdx.x * blockDim.x + threadIdx.x;
    if (e < (unsigned)E) atomicAdd(&deg[dst[e]], 1.0f);
}

__global__ void dinv_kernel(float* deg, int N) {
    unsigned i = blockIdx.x * blockDim.x + threadIdx.x;
    if (i < (unsigned)N) {
        float d = deg[i];
        deg[i] = d > 0.f ? rsqrtf(d) : 0.f;
    }
}

// ---------------- GEMM1: h1[N,16] = x[N,128] @ W1[128,16] via f32 WMMA ----------------
__global__ void gemm1_kernel(const float* __restrict__ x, const float* __restrict__ W1,
                             float* __restrict__ h1, int N) {
    int tiles = N >> 4;
    int wid = blockIdx.x * (blockDim.x >> 5) + (threadIdx.x >> 5);
    if (wid >= tiles) return;                  // wave-uniform; EXEC stays all-1s for WMMA
    int lane = threadIdx.x & 31;
    int colc = lane & 15;                      // N-col / A-row within tile
    int hi   = lane >> 4;
    int koff = hi << 1;
    int row0 = wid << 4;

    // B-layout: vgpr j of lane l holds W1[kk*4 + (l>>4)*2 + j][l&15]
    v2f wb[32];
#pragma unroll
    for (int kk = 0; kk < 32; ++kk) {
        int kr = kk * 4 + koff;
        wb[kk].x = W1[kr * HD + colc];
        wb[kk].y = W1[(kr + 1) * HD + colc];
    }
    v8f acc = {};
    const float* xr = x + (size_t)(row0 + colc) * FIN + koff;
#pragma unroll
    for (int kk = 0; kk < 32; ++kk) {
        v2f a = *(const v2f*)(xr + kk * 4);    // A-layout: contiguous pair, 8B aligned
        acc = __builtin_amdgcn_wmma_f32_16x16x4_f32(false, a, false, wb[kk],
                                                    (short)0, acc, false, false);
    }
#pragma unroll
    for (int v = 0; v < 8; ++v)
        h1[(size_t)(row0 + v + 8 * hi) * HD + colc] = acc[v];
}

// ---------------- aggregation layer 1: 16 lanes per (edge|self-loop) ----------------
__global__ void agg1_kernel(const int* __restrict__ src, const int* __restrict__ dst,
                            const float* __restrict__ dinv, const float* __restrict__ h1,
                            float* agg, int E, int N) {
    unsigned idx = blockIdx.x * blockDim.x + threadIdx.x;
    unsigned g = idx >> 4;
    if (g >= (unsigned)(E + N)) return;
    int j = idx & 15;
    int s, d;
    if (g < (unsigned)E) { s = src[g]; d = dst[g]; } else { s = d = (int)(g - E); }
    float nrm = dinv[s] * dinv[d];
    atomicAdd(&agg[(size_t)d * HD + j], h1[(size_t)s * HD + j] * nrm);
}

// ---------------- bias + relu in place, per-channel sum/sumsq (f64) ----------------
__global__ void relu_stats_kernel(float* hbuf, const float* __restrict__ b1,
                                  double* stats, int N) {
    __shared__ double ssum[16], ssq[16];
    int tid = threadIdx.x;
    if (tid < 16) { ssum[tid] = 0.0; ssq[tid] = 0.0; }
    __syncthreads();
    unsigned idx = blockIdx.x * blockDim.x + tid;
    if (idx < (unsigned)N * HD) {
        int j = idx & 15;
        float v = fmaxf(hbuf[idx] + b1[j], 0.f);
        hbuf[idx] = v;
        atomicAdd(&ssum[j], (double)v);
        atomicAdd(&ssq[j], (double)v * (double)v);
    }
    __syncthreads();
    if (tid < 16) {
        atomicAdd(&stats[tid], ssum[tid]);
        atomicAdd(&stats[16 + tid], ssq[tid]);
    }
}

__global__ void bn_finalize_kernel(const double* __restrict__ stats,
                                   const float* __restrict__ gamma,
                                   const float* __restrict__ beta,
                                   float* bn, int N) {
    int j = threadIdx.x;
    if (j < 16) {
        double mean = stats[j] / (double)N;
        double var  = stats[16 + j] / (double)N - mean * mean;   // biased var
        float sc = gamma[j] * rsqrtf((float)var + EPS);
        bn[j]      = sc;
        bn[16 + j] = beta[j] - (float)mean * sc;
    }
}

// ---------------- fused: BN affine + 10 WMMA hops + GEMM2 (h@W2) ----------------
// one wave per 16-node tile; h kept in D-layout regs; LDS roundtrip re-forms A-layout
__global__ void hops_gemm2_kernel(const float* __restrict__ hin, const float* __restrict__ bn,
                                  const float* __restrict__ Wl, const float* __restrict__ bl,
                                  const float* __restrict__ W2, float* __restrict__ c1, int N) {
    __shared__ float tile[16 * 18];            // pad stride 18: conflict-free, 8B-align irrelevant (b32)
    int lane = threadIdx.x;
    int colc = lane & 15;
    int hi   = lane >> 4;
    int koff = hi << 1;
    int row0 = blockIdx.x << 4;

    float scale = bn[colc], shift = bn[16 + colc], blv = bl[colc];
    float h[8];
#pragma unroll
    for (int v = 0; v < 8; ++v)
        h[v] = hin[(size_t)(row0 + v + 8 * hi) * HD + colc] * scale + shift;

    v2f wb[4];                                 // Wl in B-layout (constant across hops)
#pragma unroll
    for (int kk = 0; kk < 4; ++kk) {
        int kr = kk * 4 + koff;
        wb[kk].x = Wl[kr * HD + colc];
        wb[kk].y = Wl[(kr + 1) * HD + colc];
    }
    v2f w2b[12];                               // W2 in B-layout, 3 column-tiles (40 -> 48 padded)
#pragma unroll
    for (int tt = 0; tt < 3; ++tt) {
        int col = tt * 16 + colc;
#pragma unroll
        for (int kk = 0; kk < 4; ++kk) {
            int kr = kk * 4 + koff;
            w2b[tt * 4 + kk].x = (col < NC) ? W2[kr * NC + col] : 0.f;
            w2b[tt * 4 + kk].y = (col < NC) ? W2[(kr + 1) * NC + col] : 0.f;
        }
    }

    for (int hop = 0; hop < KHOPS; ++hop) {
#pragma unroll
        for (int v = 0; v < 8; ++v) tile[(v + 8 * hi) * 18 + colc] = h[v];   // D-layout out
        __syncthreads();
        v2f a[4];
#pragma unroll
        for (int kk = 0; kk < 4; ++kk) {                                     // A-layout in
            a[kk].x = tile[colc * 18 + kk * 4 + koff];
            a[kk].y = tile[colc * 18 + kk * 4 + koff + 1];
        }
        __syncthreads();
        v8f t = {};
#pragma unroll
        for (int kk = 0; kk < 4; ++kk)
            t = __builtin_amdgcn_wmma_f32_16x16x4_f32(false, a[kk], false, wb[kk],
                                                      (short)0, t, false, false);
#pragma unroll
        for (int v = 0; v < 8; ++v) {
            float u = fmaxf(t[v] + blv, 0.f);
            h[v] = AL * u + (1.0f - AL) * h[v];
        }
    }

    // GEMM2: c1[N,40] = h[N,16] @ W2[16,40]
#pragma unroll
    for (int v = 0; v < 8; ++v) tile[(v + 8 * hi) * 18 + colc] = h[v];
    __syncthreads();
    v2f a[4];
#pragma unroll
    for (int kk = 0; kk < 4; ++kk) {
        a[kk].x = tile[colc * 18 + kk * 4 + koff];
        a[kk].y = tile[colc * 18 + kk * 4 + koff + 1];
    }
#pragma unroll
    for (int tt = 0; tt < 3; ++tt) {
        v8f acc = {};
#pragma unroll
        for (int kk = 0; kk < 4; ++kk)
            acc = __builtin_amdgcn_wmma_f32_16x16x4_f32(false, a[kk], false, w2b[tt * 4 + kk],
                                                        (short)0, acc, false, false);
        int col = tt * 16 + colc;
        if (col < NC) {                        // predicated store; EXEC restored before next WMMA
#pragma unroll
            for (int v = 0; v < 8; ++v)
                c1[(size_t)(row0 + v + 8 * hi) * NC + col] = acc[v];
        }
    }
}

// ---------------- aggregation layer 2: 8 lanes x 5 features per (edge|loop) ----------------
__global__ void agg2_kernel(const int* __restrict__ src, const int* __restrict__ dst,
                            const float* __restrict__ dinv, const float* __restrict__ c1,
                            float* out, int E, int N) {
    unsigned idx = blockIdx.x * blockDim.x + threadIdx.x;
    unsigned g = idx >> 3;
    if (g >= (unsigned)(E + N)) return;
    int j = idx & 7;
    int s, d;
    if (g < (unsigned)E) { s = src[g]; d = dst[g]; } else { s = d = (int)(g - E); }
    float nrm = dinv[s] * dinv[d];
#pragma unroll
    for (int k = 0; k < 5; ++k) {
        int c = j + 8 * k;
        atomicAdd(&out[(size_t)d * NC + c], c1[(size_t)s * NC + c] * nrm);
    }
}

// ---------------- bias + log_softmax in place ----------------
__global__ void logsoftmax_kernel(float* out, const float* __restrict__ b2, int N) {
    unsigned n = blockIdx.x * blockDim.x + threadIdx.x;
    if (n >= (unsigned)N) return;
    float v[NC];
    float m = -3.0e38f;
    for (int c = 0; c < NC; ++c) {
        v[c] = out[(size_t)n * NC + c] + b2[c];
        m = fmaxf(m, v[c]);
    }
    float s = 0.f;
    for (int c = 0; c < NC; ++c) s += expf(v[c] - m);
    float lse = m + logf(s);
    for (int c = 0; c < NC; ++c) out[(size_t)n * NC + c] = v[c] - lse;
}

extern "C" void kernel_launch(void* const* d_in, const int* in_sizes, int n_in,
                              void* d_out, int out_size, void* d_ws, size_t ws_size,
                              hipStream_t stream) {
    const float* x     = (const float*)d_in[0];
    const int*   src   = (const int*)d_in[1];
    const int*   dst   = (const int*)d_in[2];
    const float* W1    = (const float*)d_in[3];
    const float* b1    = (const float*)d_in[4];
    const float* gamma = (const float*)d_in[5];
    const float* beta  = (const float*)d_in[6];
    const float* Wl    = (const float*)d_in[7];
    const float* bl    = (const float*)d_in[8];
    const float* W2    = (const float*)d_in[9];
    const float* b2    = (const float*)d_in[10];
    int N = in_sizes[0] / FIN;
    int E = in_sizes[1];
    float* out = (float*)d_out;

    // workspace layout: [stats: 32 f64][dinv: N f32][h1: N*16][h2: N*16][c1: N*40][bn: 32 f32]
    double* stats = (double*)d_ws;
    float* fb   = (float*)((char*)d_ws + 256);
    float* dinv = fb;
    float* h1   = dinv + N;
    float* h2   = h1 + (size_t)N * HD;
    float* c1   = h2 + (size_t)N * HD;
    float* bn   = c1 + (size_t)N * NC;

    int tiles = N / 16;
    const unsigned B = 256;

    init_kernel<<<((unsigned)N * NC + B - 1) / B, B, 0, stream>>>(dinv, h2, out, stats, N);
    deg_kernel<<<((unsigned)E + B - 1) / B, B, 0, stream>>>(dst, dinv, E);
    dinv_kernel<<<((unsigned)N + B - 1) / B, B, 0, stream>>>(dinv, N);
    gemm1_kernel<<<(tiles + 7) / 8, 256, 0, stream>>>(x, W1, h1, N);
    agg1_kernel<<<(((unsigned)(E + N) * 16) + B - 1) / B, B, 0, stream>>>(src, dst, dinv, h1, h2, E, N);
    relu_stats_kernel<<<((unsigned)N * HD + B - 1) / B, B, 0, stream>>>(h2, b1, stats, N);
    bn_finalize_kernel<<<1, 32, 0, stream>>>(stats, gamma, beta, bn, N);
    hops_gemm2_kernel<<<tiles, 32, 0, stream>>>(h2, bn, Wl, bl, W2, c1, N);
    agg2_kernel<<<(((unsigned)(E + N) * 8) + B - 1) / B, B, 0, stream>>>(src, dst, dinv, c1, out, E, N);
    logsoftmax_kernel<<<((unsigned)N + B - 1) / B, B, 0, stream>>>(out, b2, N);
}